// RegularGridInterpolator_51556787421875
// MI455X (gfx1250) — compile-verified
//
#include <hip/hip_runtime.h>
#include <cstdint>

// 3-D trilinear interpolation on a 256^3 regular grid (knots = linspace(0,1,256)).
//
// MI455X reasoning: values (67 MB) fits in 192 MB L2 -> gathers are L2-resident;
// compulsory HBM traffic ~131 MB -> ~5.6 us floor at 23.3 TB/s. Gather-bound, no
// matrix structure, so no WMMA. CDNA5-specific paths used instead:
//   * global_load_async_to_lds_b32 + s_wait_asynccnt to stage knot tables in LDS
//   * paired b64 gathers (adjacent corners along innermost dim)
//   * NT temporal hints on streaming coords/output so L2 stays reserved for values
//   * wave32 blocks, ds 2-addr knot lookups.

#define GRID_N 256

typedef float f2a __attribute__((ext_vector_type(2), aligned(4)));

// Returns unnormalized corner weights (wl = right-side distance, wr = left-side
// distance) and the per-dim denominator term (dl + dr); one global divide at the
// end, exactly like the reference's num/denom formulation.
__device__ __forceinline__ void bracket1d(float x, const float* __restrict__ sp,
                                          int& idx, float& wl, float& wr,
                                          float& sum) {
  int i = (int)floorf(x * (float)(GRID_N - 1));
  i = i < 0 ? 0 : (i > GRID_N - 2 ? GRID_N - 2 : i);
  float pl = sp[i];
  float pr = sp[i + 1];
  float dl = fmaxf(x - pl, 0.0f);
  float dr = fmaxf(pr - x, 0.0f);
  if ((dl == 0.0f) && (dr == 0.0f)) { dl = 1.0f; dr = 1.0f; }  // exact-hit path
  wl = dr;
  wr = dl;
  sum = dl + dr;
  idx = i;
}

extern "C" __global__ void __launch_bounds__(256)
trilerp256_kernel(const float* __restrict__ x0, const float* __restrict__ x1,
                  const float* __restrict__ x2, const float* __restrict__ p0,
                  const float* __restrict__ p1, const float* __restrict__ p2,
                  const float* __restrict__ values, float* __restrict__ out,
                  int nq) {
  __shared__ float sp[3 * GRID_N];
  const int t = threadIdx.x;

  // ---- Stage the three 256-entry knot tables into LDS with CDNA5 async
  // global->LDS DMA (ASYNCcnt-tracked). blockDim.x == 256 == GRID_N, so each
  // lane copies exactly one element per table. The low 32 bits of a generic
  // pointer into LDS are the wave-relative LDS byte address (ISA 10.2).
  {
    unsigned lds0 = (unsigned)(uintptr_t)(&sp[0]) + (unsigned)(t * 4);
    unsigned lds1 = lds0 + GRID_N * 4u;
    unsigned lds2 = lds1 + GRID_N * 4u;
    unsigned goff = (unsigned)(t * 4);
    asm volatile("global_load_async_to_lds_b32 %0, %1, %2"
                 :: "v"(lds0), "v"(goff), "s"(p0) : "memory");
    asm volatile("global_load_async_to_lds_b32 %0, %1, %2"
                 :: "v"(lds1), "v"(goff), "s"(p1) : "memory");
    asm volatile("global_load_async_to_lds_b32 %0, %1, %2"
                 :: "v"(lds2), "v"(goff), "s"(p2) : "memory");
    asm volatile("s_wait_asynccnt 0" ::: "memory");
  }
  __syncthreads();

  const int gid = blockIdx.x * 256 + t;
  if (gid >= nq) return;

  // Streaming, touched-once operands: non-temporal so they do not evict the
  // L2-resident values array.
  const float X0 = __builtin_nontemporal_load(x0 + gid);
  const float X1 = __builtin_nontemporal_load(x1 + gid);
  const float X2 = __builtin_nontemporal_load(x2 + gid);

  int i0, i1, i2;
  float w0l, w0r, w1l, w1r, w2l, w2r, s0, s1, s2;
  bracket1d(X0, &sp[0],          i0, w0l, w0r, s0);
  bracket1d(X1, &sp[GRID_N],     i1, w1l, w1r, s1);
  bracket1d(X2, &sp[2 * GRID_N], i2, w2l, w2r, s2);

  // Flat base index fits in 32-bit (< 2^24).
  const int base = ((i0 * GRID_N) + i1) * GRID_N + i2;
  const float* b = values + base;

  // Corners along innermost dim are adjacent -> 4 b64 gathers instead of 8 b32.
  // Regular temporal hint: these must stay cached in L2.
  f2a v00 = *(const f2a*)(b);
  f2a v01 = *(const f2a*)(b + GRID_N);
  f2a v10 = *(const f2a*)(b + GRID_N * GRID_N);
  f2a v11 = *(const f2a*)(b + GRID_N * GRID_N + GRID_N);

  float c00 = v00.x * w2l + v00.y * w2r;
  float c01 = v01.x * w2l + v01.y * w2r;
  float c10 = v10.x * w2l + v10.y * w2r;
  float c11 = v11.x * w2l + v11.y * w2r;

  float c0 = c00 * w1l + c01 * w1r;
  float c1 = c10 * w1l + c11 * w1r;

  float num = c0 * w0l + c1 * w0r;
  float result = num / (s0 * s1 * s2);  // single divide, like the reference

  __builtin_nontemporal_store(result, out + gid);
}

extern "C" void kernel_launch(void* const* d_in, const int* in_sizes, int n_in,
                              void* d_out, int out_size, void* d_ws, size_t ws_size,
                              hipStream_t stream) {
  const float* x0 = (const float*)d_in[0];
  const float* x1 = (const float*)d_in[1];
  const float* x2 = (const float*)d_in[2];
  const float* p0 = (const float*)d_in[3];
  const float* p1 = (const float*)d_in[4];
  const float* p2 = (const float*)d_in[5];
  const float* values = (const float*)d_in[6];
  float* out = (float*)d_out;

  const int nq = in_sizes[0];
  const int blocks = (nq + 255) / 256;
  hipLaunchKernelGGL(trilerp256_kernel, dim3(blocks), dim3(256), 0, stream,
                     x0, x1, x2, p0, p1, p2, values, out, nq);
}